// VInput_82446192214214
// MI455X (gfx1250) — compile-verified
//
#include <hip/hip_runtime.h>

// CDNA5 / gfx1250: wave32, WMMA 16x16x4 f32.
typedef __attribute__((ext_vector_type(2))) float v2f;
typedef __attribute__((ext_vector_type(8))) float v8f;

// Problem constants from the reference.
#define BB 2
#define TT 2048
#define DD 127
#define UU 128

// out[b,t,i,j] = x[b,t,i] * coeffs[(i+j)&127],  x[...,127] = 1.0 (bias col).
// Formulated as rank-1 outer products computed with V_WMMA_F32_16X16X4_F32:
//   A (16x4): column K=0 holds x[tbase+0..15, i]   (lanes 0-15, VGPR0)
//   B (4x16): row    K=0 holds coeffs[(i+jbase+n)&127] (lanes 0-15, VGPR0)
//   D (16x16) = tile of out[t, i, j].
__global__ __launch_bounds__(256) void circ_outer_wmma(
    const float* __restrict__ inp,     // [B, T, D] f32
    const float* __restrict__ coeffs,  // [U] f32
    float* __restrict__ out)           // [B, T, U, U] f32
{
    __shared__ float sc[UU];
    const int tid = threadIdx.x;
    if (tid < UU) sc[tid] = coeffs[tid];
    __syncthreads();

    const int lane = tid & 31;
    const int waveInBlk = tid >> 5;
    const int w = blockIdx.x * 8 + waveInBlk;  // 32768 wave-tasks total

    const int i  = w & 127;          // row index within D1=128
    const int tt = (w >> 7) & 127;   // t-tile (T/16 = 128)
    const int b  = w >> 14;          // batch

    const int tbase  = tt * 16;
    const int col    = lane & 15;        // N (j within tile) / M for A load
    const int rowOff = (lane >> 4) * 8;  // C/D: lanes 16-31 hold m = r + 8

    // ---- A matrix: x[tbase+m, i] in K=0 column (lanes 0-15 only) ----
    float xv = 1.0f;                               // bias column i == 127
    if (i < DD) {
        xv = inp[((size_t)(b * TT + tbase + col)) * DD + i];
    }
    v2f a;
    a.x = (lane < 16) ? xv : 0.0f;  // K=0 (lanes 0-15) ; K=2 (lanes 16-31) = 0
    a.y = 0.0f;                     // K=1 / K=3 = 0

    // Per-lane output base: out[b, tbase+rowOff, i, col]
    float* p = out + ((size_t)(b * TT + tbase + rowOff) * UU + i) * UU + col;

    #pragma unroll
    for (int J = 0; J < 8; ++J) {
        const int jbase = J * 16;

        // ---- B matrix: rotated coeffs row in K=0 (lanes 0-15 only) ----
        const float wv = sc[(i + jbase + col) & 127];
        v2f bb;
        bb.x = (lane < 16) ? wv : 0.0f;
        bb.y = 0.0f;

        v8f c = {};
        // 8 args: (neg_a, A, neg_b, B, c_mod, C, reuse_a, reuse_b)
        v8f d = __builtin_amdgcn_wmma_f32_16x16x4_f32(
            false, a, false, bb, (short)0, c, false, false);

        // D VGPR r: lanes 0-15 -> (m=r, n=lane); lanes 16-31 -> (m=r+8, n=lane-16)
        float* pj = p + jbase;
        #pragma unroll
        for (int r = 0; r < 8; ++r) {
            pj[(size_t)r * (UU * UU)] = d[r];  // t-stride = 128*128 floats
        }
    }
}

extern "C" void kernel_launch(void* const* d_in, const int* in_sizes, int n_in,
                              void* d_out, int out_size, void* d_ws, size_t ws_size,
                              hipStream_t stream) {
    const float* inp = (const float*)d_in[0];   // [2, 2048, 127] f32
    const float* cf  = (const float*)d_in[1];   // [128] f32
    float* out = (float*)d_out;                 // [2, 2048, 128, 128] f32

    // 32768 wave-tasks, 8 waves (256 threads) per block -> 4096 blocks.
    hipLaunchKernelGGL(circ_outer_wmma, dim3(4096), dim3(256), 0, stream,
                       inp, cf, out);
}